// _ComplexStateSpaceModel_72980084294159
// MI455X (gfx1250) — compile-verified
//
#include <hip/hip_runtime.h>

// ---------------- problem constants ----------------
#define BATCH   8
#define TLEN    2048
#define INDIM   128     // input feature dim
#define STDIM   256     // state dim (== threads per block, 1 state per thread)
#define OUTD    128     // output dim
#define CT      32      // timesteps per chunk (also the GEMM M dimension)
#define NC      (TLEN / CT)     // 64 chunks -> BATCH*NC = 512 workgroups
#define NTHREADS 256            // 8 waves (wave32)

typedef __attribute__((ext_vector_type(2))) float v2f;
typedef __attribute__((ext_vector_type(8))) float v8f;

// CDNA5 fp32 WMMA: D[16x16] = A[16x4] * B[4x16] + C   (v_wmma_f32_16x16x4_f32)
__device__ __forceinline__ v8f wmma_f32(v2f a, v2f b, v8f c) {
  return __builtin_amdgcn_wmma_f32_16x16x4_f32(false, a, false, b, (short)0, c,
                                               false, false);
}

// ---------------- pass 0: stabilize A, compute A^CT ----------------
__global__ void ssm_prep(const float* __restrict__ Are,
                         const float* __restrict__ Aim,
                         float* __restrict__ ws_a,    // [2][STDIM] re,im
                         float* __restrict__ ws_ap) { // [2][STDIM] re,im of a^CT
  const int n = threadIdx.x;
  const float re = Are[n], im = Aim[n];
  // a = A / sqrt(|A|^2 + 1)
  const float inv = rsqrtf(re * re + im * im + 1.0f);
  const float ar = re * inv, ai = im * inv;
  ws_a[n] = ar;
  ws_a[STDIM + n] = ai;
  float pr = ar, pi = ai;
  for (int i = 1; i < CT; i <<= 1) {   // log2(CT) squarings -> a^CT
    const float nr = pr * pr - pi * pi;
    const float ni = 2.0f * pr * pi;
    pr = nr; pi = ni;
  }
  ws_ap[n] = pr;
  ws_ap[STDIM + n] = pi;
}

// ---------------- pass 1/3: fused (u@B) GEMM + diagonal scan (+ y GEMM) ------
// FINAL=false: zero-init local scan per chunk, emit end state e[b][c][:].
// FINAL=true : scan with carry g[b][c][:] init, emit y = Re(x@C) + u@D + b.
template <bool FINAL>
__global__ __launch_bounds__(NTHREADS) void ssm_chunk(
    const float* __restrict__ u,
    const float* __restrict__ Bre, const float* __restrict__ Bim,
    const float* __restrict__ Cre, const float* __restrict__ Cim,
    const float* __restrict__ Dm,  const float* __restrict__ bias,
    const float* __restrict__ ws_a,
    float* __restrict__ ws_e,        // [B][NC][STDIM][2]
    const float* __restrict__ ws_g,  // [B][NC][STDIM][2]
    float* __restrict__ out_y) {
  __shared__ float lds_u[CT * INDIM];   // u chunk      [time][in]    16 KB
  __shared__ float lds_re[CT * STDIM];  // uB_re / x_re [time][state] 32 KB
  __shared__ float lds_im[CT * STDIM];  // uB_im / x_im [time][state] 32 KB

  const int bb = blockIdx.x / NC;
  const int cc = blockIdx.x % NC;
  const int tid  = threadIdx.x;
  const int wave = tid >> 5;
  const int lane = tid & 31;
  const int half = lane >> 4;   // half-wave -> K split of fp32 WMMA fragments
  const int l16  = lane & 15;   // row (A) / column (B,C,D) within a 16-tile
  const int tbase = cc * CT;

  // Warm L2/L0 for the weight matrices while the u tile streams in.
  for (int off = tid * 64; off < INDIM * STDIM; off += NTHREADS * 64) {
    __builtin_prefetch(Bre + off, 0, 3);
    __builtin_prefetch(Bim + off, 0, 3);
  }
  if (FINAL) {
    for (int off = tid * 64; off < STDIM * OUTD; off += NTHREADS * 64) {
      __builtin_prefetch(Cre + off, 0, 3);
      __builtin_prefetch(Cim + off, 0, 3);
    }
  }

  // ---- load u chunk [CT x INDIM] as b128 (coalesced) ----
  {
    const float4* usrc =
        (const float4*)(u + (size_t)(bb * TLEN + tbase) * INDIM);
    float4* udst = (float4*)lds_u;
    for (int i = tid; i < CT * INDIM / 4; i += NTHREADS) udst[i] = usrc[i];
  }
  __syncthreads();

  // ---- GEMM1: uB[CT x STDIM] = u_chunk @ {Bre,Bim} ----
  // 2 M-tiles (rows 0-15 / 16-31) share every B fragment; each wave owns 4 of
  // the 32 16-wide tile-columns (re plane: cols 0..15, im plane: cols 16..31).
  {
    const float* Bp[4]  = {Bre, Bre, Bim, Bim};
    const int    n0q[4] = {wave * 16, (wave + 8) * 16, wave * 16,
                           (wave + 8) * 16};
    v8f acc[2][4] = {};
    for (int k0 = 0; k0 < INDIM; k0 += 4) {
      const int kk = k0 + half * 2;
      const v2f av0 = *(const v2f*)(lds_u + l16 * INDIM + kk);
      const v2f av1 = *(const v2f*)(lds_u + (l16 + 16) * INDIM + kk);
#pragma unroll
      for (int q = 0; q < 4; ++q) {
        v2f bv;
        bv.x = Bp[q][(size_t)(kk + 0) * STDIM + n0q[q] + l16];
        bv.y = Bp[q][(size_t)(kk + 1) * STDIM + n0q[q] + l16];
        acc[0][q] = wmma_f32(av0, bv, acc[0][q]);
        acc[1][q] = wmma_f32(av1, bv, acc[1][q]);
      }
    }
#pragma unroll
    for (int mt = 0; mt < 2; ++mt) {
#pragma unroll
      for (int q = 0; q < 4; ++q) {
        float* plane = (q < 2) ? lds_re : lds_im;
#pragma unroll
        for (int r = 0; r < 8; ++r)
          plane[(mt * 16 + r + 8 * half) * STDIM + n0q[q] + l16] =
              acc[mt][q][r];
      }
    }
  }
  __syncthreads();

  // ---- diagonal complex scan over the CT timesteps (thread = state col) ----
  {
    const float ar = ws_a[tid];
    const float ai = ws_a[STDIM + tid];
    float xr, xi;
    if (FINAL) {
      const size_t gi = ((size_t)(bb * NC + cc) * STDIM + tid) * 2;
      xr = ws_g[gi + 0];
      xi = ws_g[gi + 1];
    } else {
      xr = 0.0f; xi = 0.0f;
    }
#pragma unroll
    for (int j = 0; j < CT; ++j) {
      const float br = lds_re[j * STDIM + tid];
      const float bi = lds_im[j * STDIM + tid];
      const float nr = ar * xr - ai * xi + br;
      const float ni = ar * xi + ai * xr + bi;
      xr = nr; xi = ni;
      if (FINAL) {  // overwrite uB planes with x (same address per thread)
        lds_re[j * STDIM + tid] = xr;
        lds_im[j * STDIM + tid] = xi;
      }
    }
    if (!FINAL) {
      const size_t ei = ((size_t)(bb * NC + cc) * STDIM + tid) * 2;
      ws_e[ei + 0] = xr;
      ws_e[ei + 1] = xi;
    }
  }

  if (FINAL) {
    __syncthreads();
    // ---- GEMM2: y[CT x OUTD] = x_re@Cre + x_im@(-Cim) + u@D + bias ----
    // wave owns output columns n0..n0+15; both M-tiles share each C/D frag.
    const int n0 = wave * 16;
    v8f acc[2] = {};
    for (int k0 = 0; k0 < STDIM; k0 += 4) {
      const int kk = k0 + half * 2;
      const v2f av0 = *(const v2f*)(lds_re + l16 * STDIM + kk);
      const v2f av1 = *(const v2f*)(lds_re + (l16 + 16) * STDIM + kk);
      v2f bv;
      bv.x = Cre[(size_t)(kk + 0) * OUTD + n0 + l16];
      bv.y = Cre[(size_t)(kk + 1) * OUTD + n0 + l16];
      acc[0] = wmma_f32(av0, bv, acc[0]);
      acc[1] = wmma_f32(av1, bv, acc[1]);
    }
    for (int k0 = 0; k0 < STDIM; k0 += 4) {
      const int kk = k0 + half * 2;
      const v2f av0 = *(const v2f*)(lds_im + l16 * STDIM + kk);
      const v2f av1 = *(const v2f*)(lds_im + (l16 + 16) * STDIM + kk);
      v2f bv;  // fold the minus sign of Re(x@C) into the B fragment
      bv.x = -Cim[(size_t)(kk + 0) * OUTD + n0 + l16];
      bv.y = -Cim[(size_t)(kk + 1) * OUTD + n0 + l16];
      acc[0] = wmma_f32(av0, bv, acc[0]);
      acc[1] = wmma_f32(av1, bv, acc[1]);
    }
    for (int k0 = 0; k0 < INDIM; k0 += 4) {
      const int kk = k0 + half * 2;
      const v2f av0 = *(const v2f*)(lds_u + l16 * INDIM + kk);
      const v2f av1 = *(const v2f*)(lds_u + (l16 + 16) * INDIM + kk);
      v2f bv;
      bv.x = Dm[(size_t)(kk + 0) * OUTD + n0 + l16];
      bv.y = Dm[(size_t)(kk + 1) * OUTD + n0 + l16];
      acc[0] = wmma_f32(av0, bv, acc[0]);
      acc[1] = wmma_f32(av1, bv, acc[1]);
    }
    const float bz = bias[n0 + l16];
#pragma unroll
    for (int mt = 0; mt < 2; ++mt) {
#pragma unroll
      for (int r = 0; r < 8; ++r) {
        const int m = mt * 16 + r + 8 * half;
        out_y[(size_t)(bb * TLEN + tbase + m) * OUTD + n0 + l16] =
            acc[mt][r] + bz;
      }
    }
  }
}

// ---------------- pass 2: combine chunk carries g[c+1] = A^CT*g[c] + e[c] ----
__global__ void ssm_carry(const float* __restrict__ x0re,
                          const float* __restrict__ x0im,
                          const float* __restrict__ ws_ap,
                          const float* __restrict__ ws_e,
                          float* __restrict__ ws_g,
                          float* __restrict__ out_x) {  // final state, complex64
  const int bb = blockIdx.x;
  const int n  = threadIdx.x;
  const float pr = ws_ap[n], pi = ws_ap[STDIM + n];
  float gr = x0re[bb * STDIM + n];
  float gi = x0im[bb * STDIM + n];
  for (int c = 0; c < NC; ++c) {
    const size_t idx = ((size_t)(bb * NC + c) * STDIM + n) * 2;
    ws_g[idx + 0] = gr;
    ws_g[idx + 1] = gi;
    const float er = ws_e[idx + 0];
    const float ei = ws_e[idx + 1];
    const float nr = pr * gr - pi * gi + er;
    const float ni = pr * gi + pi * gr + ei;
    gr = nr; gi = ni;
  }
  out_x[((size_t)bb * STDIM + n) * 2 + 0] = gr;
  out_x[((size_t)bb * STDIM + n) * 2 + 1] = gi;
}

// ---------------- host-side launcher ----------------
extern "C" void kernel_launch(void* const* d_in, const int* in_sizes, int n_in,
                              void* d_out, int out_size, void* d_ws,
                              size_t ws_size, hipStream_t stream) {
  const float* u    = (const float*)d_in[0];
  const float* x0re = (const float*)d_in[1];
  const float* x0im = (const float*)d_in[2];
  const float* Are  = (const float*)d_in[3];
  const float* Aim  = (const float*)d_in[4];
  const float* Bre  = (const float*)d_in[5];
  const float* Bim  = (const float*)d_in[6];
  const float* Cre  = (const float*)d_in[7];
  const float* Cim  = (const float*)d_in[8];
  const float* Dm   = (const float*)d_in[9];
  const float* bias = (const float*)d_in[10];
  float* out = (float*)d_out;
  float* ws  = (float*)d_ws;

  float* ws_a  = ws;                 // [2][STDIM]
  float* ws_ap = ws + 2 * STDIM;     // [2][STDIM]
  float* ws_e  = ws + 4 * STDIM;     // [B][NC][STDIM][2]
  float* ws_g  = ws_e + (size_t)BATCH * NC * STDIM * 2;
  float* out_y = out;                                   // [B][T][OUTD] f32
  float* out_x = out + (size_t)BATCH * TLEN * OUTD;     // [B][STDIM] c64

  ssm_prep<<<1, STDIM, 0, stream>>>(Are, Aim, ws_a, ws_ap);
  ssm_chunk<false><<<BATCH * NC, NTHREADS, 0, stream>>>(
      u, Bre, Bim, Cre, Cim, Dm, bias, ws_a, ws_e, ws_g, out_y);
  ssm_carry<<<BATCH, STDIM, 0, stream>>>(x0re, x0im, ws_ap, ws_e, ws_g, out_x);
  ssm_chunk<true><<<BATCH * NC, NTHREADS, 0, stream>>>(
      u, Bre, Bim, Cre, Cim, Dm, bias, ws_a, ws_e, ws_g, out_y);
}